// PhotonicRecurrentNeuron_61100204753058
// MI455X (gfx1250) — compile-verified
//
#include <hip/hip_runtime.h>
#include <stdint.h>

typedef float v2f __attribute__((ext_vector_type(2)));
typedef float v8f __attribute__((ext_vector_type(8)));

#define T_STEPS 256
#define BATCH   4096
#define IDIM    144
#define HDIM    32
#define KDIM    (HDIM + IDIM)   // 176

#define ROWS_PER_BLOCK 16   // one wave-pair per block: both waves work on the same 16 rows
#define THREADS        64   // 2 waves: s=0 -> k-steps 0..21, s=1 -> k-steps 22..43
#define XPITCH         148  // padded x-tile row pitch (floats): conflict-free A-frag reads
#define HPITCH         36   // padded h/partial-tile row pitch (floats)
#define KSTEPS         44   // K=176 in steps of 4
#define KSLICE         22   // k-steps per wave
#define KSTEPS_H       8    // h occupies k-steps 0..7 (entirely in slice 0)
#define COPIES         9    // float4 async copies per thread per step: 16*36/64

__device__ __forceinline__ float psig(float z) {
  // A2 + (A1-A2) / (1 + exp(min(z - X0, CUTOFF)/D))
  float zz = fminf(z - 0.145f, 2.0f);
  return 1.005f + (0.06f - 1.005f) / (1.0f + __expf(zz * (1.0f / 0.033f)));
}

__global__ __launch_bounds__(THREADS)
void prn_kernel(const float* __restrict__ x, const float* __restrict__ W,
                const float* __restrict__ b, float* __restrict__ out) {
  __shared__ __align__(16) float Xls[2 * ROWS_PER_BLOCK * XPITCH]; // double-buffered x tile
  __shared__ __align__(16) float Hls[ROWS_PER_BLOCK * HPITCH];     // recurrent state tile
  __shared__ __align__(16) float Zls[ROWS_PER_BLOCK * HPITCH];     // slice-1 partial sums (C-layout lanes)

  const int tid  = threadIdx.x;
  const int s    = tid >> 5;    // K-slice of this wave (0 or 1)
  const int lane = tid & 31;
  const int n0   = lane & 15;   // N within 16-wide tile
  const int g    = lane >> 4;   // lane half-group
  const int koff = 2 * g;       // K offset of this lane's A/B fragment pair
  const int row0 = blockIdx.x * ROWS_PER_BLOCK;
  const int ksbase = s * KSLICE;

  // ---- hoist this wave's K-slice of W B-fragments into registers (88 VGPRs) ----
  // wreg[j*2+nt] = { Wcat[K=4(ksbase+j)+2g][N=n0+16nt], ...+1 },  Wcat[k][n] = W[n][k]
  v2f wreg[KSLICE * 2];
#pragma unroll
  for (int j = 0; j < KSLICE; ++j) {
#pragma unroll
    for (int nt = 0; nt < 2; ++nt)
      wreg[j * 2 + nt] = *(const v2f*)&W[(n0 + 16 * nt) * KDIM + (ksbase + j) * 4 + koff];
  }

  // ---- h0 = 0 ----
  for (int idx = tid; idx < ROWS_PER_BLOCK * HPITCH; idx += THREADS)
    Hls[idx] = 0.0f;

  // bias splat (added once, by slice 0)
  const float b0v = b[n0];
  const float b1v = b[n0 + 16];

  // ---- precompute async-copy offsets (flat float4 index -> row/col of x tile) ----
  uint32_t goff[COPIES], loff[COPIES];
#pragma unroll
  for (int j = 0; j < COPIES; ++j) {
    int f4 = tid + j * THREADS;
    int r  = f4 / 36;           // 36 float4 per 144-float row
    int c  = f4 - r * 36;
    goff[j] = (uint32_t)(r * (IDIM * 4) + c * 16);
    loff[j] = (uint32_t)(uintptr_t)(&Xls[r * XPITCH + c * 4]); // low 32 bits = LDS offset
  }
  const uint32_t xbuf_bytes = (uint32_t)(ROWS_PER_BLOCK * XPITCH * 4);

  __syncthreads();

  // prologue: async-stage x slice t=0 into buffer 0
  {
    const float* src = x + (size_t)row0 * IDIM;
#pragma unroll
    for (int j = 0; j < COPIES; ++j)
      asm volatile("global_load_async_to_lds_b128 %0, %1, %2"
                   :: "v"(loff[j]), "v"(goff[j]), "s"(src) : "memory");
  }

  const int hrow = n0 * HPITCH;
  const int xrow = n0 * XPITCH;

  for (int t = 0; t < T_STEPS; ++t) {
    // issue async copy for next step (index clamped; duplicate last slice is harmless)
    {
      int tn = (t + 1 < T_STEPS) ? (t + 1) : (T_STEPS - 1);
      uint32_t pbase = ((t + 1) & 1) ? xbuf_bytes : 0u;
      const float* src = x + ((size_t)tn * BATCH + row0) * IDIM;
#pragma unroll
      for (int j = 0; j < COPIES; ++j)
        asm volatile("global_load_async_to_lds_b128 %0, %1, %2"
                     :: "v"(loff[j] + pbase), "v"(goff[j]), "s"(src) : "memory");
    }
    // async loads complete in order: waiting to <=9 guarantees this step's 9 are done
    asm volatile("s_wait_asynccnt 9" ::: "memory");
    __syncthreads();

    const float* Xb = &Xls[(t & 1) ? (ROWS_PER_BLOCK * XPITCH) : 0];

    // ---- burst-load this slice's 22 A-fragments into registers ----
    v2f a[KSLICE];
    if (s == 0) {
#pragma unroll
      for (int j = 0; j < KSTEPS_H; ++j)                 // k-steps 0..7: recurrent state
        a[j] = *(const v2f*)&Hls[hrow + j * 4 + koff];
#pragma unroll
      for (int j = 0; j < KSLICE - KSTEPS_H; ++j)        // k-steps 8..21: x cols 0..55
        a[KSTEPS_H + j] = *(const v2f*)&Xb[xrow + j * 4 + koff];
    } else {
#pragma unroll
      for (int j = 0; j < KSLICE; ++j)                   // k-steps 22..43: x cols 56..143
        a[j] = *(const v2f*)&Xb[xrow + (j + 14) * 4 + koff];
    }

    v8f acc0, acc1;
    const float i0 = (s == 0) ? b0v : 0.0f;
    const float i1 = (s == 0) ? b1v : 0.0f;
#pragma unroll
    for (int v = 0; v < 8; ++v) { acc0[v] = i0; acc1[v] = i1; }

    // ---- 44 chained f32 WMMAs over this K-slice ----
#pragma unroll
    for (int j = 0; j < KSLICE; ++j) {
      acc0 = __builtin_amdgcn_wmma_f32_16x16x4_f32(false, a[j], false, wreg[j * 2 + 0], (short)0, acc0, false, false);
      acc1 = __builtin_amdgcn_wmma_f32_16x16x4_f32(false, a[j], false, wreg[j * 2 + 1], (short)0, acc1, false, false);
    }

    // ---- slice 1 publishes partial sums (C-layout: lane->(m,n) identical across waves) ----
    if (s == 1) {
#pragma unroll
      for (int v = 0; v < 8; ++v) {
        int m = 8 * g + v;
        Zls[m * HPITCH + n0]      = acc0[v];
        Zls[m * HPITCH + 16 + n0] = acc1[v];
      }
    }
    __syncthreads();

    // ---- slice 0 reduces, applies activation, updates recurrent state ----
    if (s == 0) {
#pragma unroll
      for (int v = 0; v < 8; ++v) {
        int m = 8 * g + v;
        float z0 = acc0[v] + Zls[m * HPITCH + n0];
        float z1 = acc1[v] + Zls[m * HPITCH + 16 + n0];
        Hls[m * HPITCH + n0]      = psig(z0);
        Hls[m * HPITCH + 16 + n0] = psig(z1);
      }
    }
    __syncthreads();  // h + partials consumed; x buffer free for refill
  }

  // ---- final h -> global (coalesced, sourced from LDS) ----
  for (int idx = tid; idx < ROWS_PER_BLOCK * HDIM; idx += THREADS) {
    int r = idx >> 5;
    int c = idx & 31;
    out[(size_t)(row0 + r) * HDIM + c] = Hls[r * HPITCH + c];
  }
}

extern "C" void kernel_launch(void* const* d_in, const int* in_sizes, int n_in,
                              void* d_out, int out_size, void* d_ws, size_t ws_size,
                              hipStream_t stream) {
  const float* x = (const float*)d_in[0];
  const float* W = (const float*)d_in[1];
  const float* b = (const float*)d_in[2];
  float* out = (float*)d_out;
  (void)in_sizes; (void)n_in; (void)out_size; (void)d_ws; (void)ws_size;
  dim3 grid(BATCH / ROWS_PER_BLOCK);
  dim3 block(THREADS);
  hipLaunchKernelGGL(prn_kernel, grid, block, 0, stream, x, W, b, out);
}